// GraphAttentionLayer_68710886801773
// MI455X (gfx1250) — compile-verified
//
#include <hip/hip_runtime.h>
#include <hip/hip_bf16.h>
#include <math.h>

#define ALPHA 0.2f
#define MAXNB 256

typedef __attribute__((ext_vector_type(2))) float v2f;
typedef __attribute__((ext_vector_type(8))) float v8f;

// ---------------------------------------------------------------------------
// Phase 0: [p1|p2] = X @ [a1|a2] via V_WMMA_F32_16X16X4_F32.
// One wave per 16-row tile; 32 K-steps of K=4 sweep the F=128 reduction.
// A-operand (16x4 fp32): lanes 0-15 hold K=k0,k0+1 for row M=lane&15;
// lanes 16-31 hold K=k0+2,k0+3. B-operand (4x16): col=lane&15, K rows
// k0+2*half,+1. B has col0=a1, col1=a2, rest zero — pre-baked per (step,lane)
// into an LDS table so the inner loop is branch-free (1 ds_load_b64/step).
// D (16x16 f32): VGPR r = row half*8+r, col = lane&15. Col0 = X·a1, col1 = X·a2.
// mode 0 (query): qa1[row] = col0; selfe[row] = lrelu(col0+col1)
// mode 1 (key)  : ka2[row] = col1
// ---------------------------------------------------------------------------
__global__ void __launch_bounds__(128)
gat_proj_wmma(const int* __restrict__ ver, int expect,
              const float* __restrict__ X, int N,
              const float* __restrict__ a, int mode,
              float* __restrict__ qa1,
              float* __restrict__ selfe,
              float* __restrict__ ka2) {
    if (*ver != expect) return;

    // s_b[s][lane][2] : the exact v2f B operand lane `lane` feeds at step s.
    __shared__ float s_b[32 * 32 * 2];   // 8 KB
    for (int t = threadIdx.x; t < 2048; t += 128) {
        const int s     = t >> 6;        // K-step
        const int l     = (t >> 1) & 31; // lane
        const int which = t & 1;         // element 0/1 of the pair
        const int half  = l >> 4;
        const int col   = l & 15;
        const int k     = 4 * s + 2 * half + which;
        float v = 0.0f;
        if (col == 0)      v = a[k];        // column 0 <- a1
        else if (col == 1) v = a[128 + k];  // column 1 <- a2
        s_b[t] = v;
    }
    __syncthreads();

    const int wave  = threadIdx.x >> 5;
    const int lane  = threadIdx.x & 31;
    const int tiles = (N + 15) >> 4;
    const int tile  = blockIdx.x * 4 + wave;
    if (tile >= tiles) return;

    const int r0   = tile << 4;
    const int half = lane >> 4;
    const int col  = lane & 15;
    int arow = r0 + col;
    if (arow >= N) arow = 0;   // clamp: garbage lands in D rows >= N, never stored

    const v2f* __restrict__ Arow = (const v2f*)(X + (size_t)arow * 128);
    const v2f* __restrict__ Bp   = (const v2f*)s_b;

    // Two accumulator chains over even/odd steps to overlap WMMA latency.
    v8f c0 = {}, c1 = {};
#pragma unroll
    for (int s = 0; s < 32; s += 2) {
        v2f a0 = Arow[2 * s + half];
        v2f b0 = Bp[s * 32 + lane];
        c0 = __builtin_amdgcn_wmma_f32_16x16x4_f32(false, a0, false, b0,
                                                   (short)0, c0, false, false);
        v2f a1 = Arow[2 * (s + 1) + half];
        v2f b1 = Bp[(s + 1) * 32 + lane];
        c1 = __builtin_amdgcn_wmma_f32_16x16x4_f32(false, a1, false, b1,
                                                   (short)0, c1, false, false);
    }

    // Extract columns 0 (X·a1) and 1 (X·a2) of the D tile.
#pragma unroll
    for (int r = 0; r < 8; ++r) {
        const float cr = c0[r] + c1[r];
        const float v1 = __shfl(cr, half * 16 + 0, 32);  // col 0, this half
        const float v2 = __shfl(cr, half * 16 + 1, 32);  // col 1, this half
        if (col == 0) {
            const int orow = r0 + half * 8 + r;
            if (orow < N) {
                if (mode == 0) {
                    qa1[orow] = v1;
                    const float se = v1 + v2;
                    selfe[orow] = (se > 0.0f) ? se : ALPHA * se;
                } else {
                    ka2[orow] = v2;
                }
            }
        }
    }
}

// ---------------------------------------------------------------------------
// Phase 1: one 128-thread block per output row.
//  - float4 streaming scan of the dense adj row, append nonzeros to LDS
//  - block-reduce max & sum(exp) incl. the always-present self edge
//  - gather-accumulate neighbor rows of Uk, elu, store 128 floats
// ---------------------------------------------------------------------------
__global__ void __launch_bounds__(128)
gat_row(const int* __restrict__ ver, int expect,
        const float* __restrict__ Uq, const float* __restrict__ Uk,
        const float* __restrict__ adj,
        const float* __restrict__ qa1, const float* __restrict__ selfe,
        const float* __restrict__ ka2,
        int Nq, int Nk, float* __restrict__ out) {
    if (*ver != expect) return;

    const int i = blockIdx.x;
    if (i >= Nq) return;
    const int tid = threadIdx.x;

    __shared__ int   s_idx[MAXNB];
    __shared__ float s_e[MAXNB];
    __shared__ float s_v[MAXNB];
    __shared__ float s_red[128];
    __shared__ int   s_cnt;

    if (tid == 0) s_cnt = 0;
    __syncthreads();

    const float qa   = qa1[i];
    const float se_i = selfe[i];

    // ---- scan adjacency row (Nk divisible by 4; rows are 16B aligned) ----
    const float4* __restrict__ rowp =
        (const float4*)(adj + (size_t)i * (size_t)Nk);
    const int nq4 = Nk >> 2;
    for (int q = tid; q < nq4; q += 128) {
        float4 v4 = rowp[q];
        float vv[4] = {v4.x, v4.y, v4.z, v4.w};
#pragma unroll
        for (int cidx = 0; cidx < 4; ++cidx) {
            float v = vv[cidx];
            if (v > 0.0f) {
                int j = q * 4 + cidx;
                float e = qa + ka2[j];
                e = (e > 0.0f) ? e : ALPHA * e;
                int p = atomicAdd(&s_cnt, 1);
                if (p < MAXNB) { s_idx[p] = j; s_e[p] = e; s_v[p] = v; }
            }
        }
    }
    __syncthreads();

    int cnt = s_cnt;
    if (cnt > MAXNB) cnt = MAXNB;

    // ---- max reduce (self edge always present) ----
    float lm = -3.0e38f;
    for (int k = tid; k < cnt; k += 128) lm = fmaxf(lm, s_e[k]);
    s_red[tid] = lm;
    __syncthreads();
#pragma unroll
    for (int off = 64; off > 0; off >>= 1) {
        if (tid < off) s_red[tid] = fmaxf(s_red[tid], s_red[tid + off]);
        __syncthreads();
    }
    const float m = fmaxf(s_red[0], se_i);
    __syncthreads();

    // ---- sum of exp ----
    float ls = 0.0f;
    for (int k = tid; k < cnt; k += 128) ls += expf(s_e[k] - m);
    s_red[tid] = ls;
    __syncthreads();
#pragma unroll
    for (int off = 64; off > 0; off >>= 1) {
        if (tid < off) s_red[tid] += s_red[tid + off];
        __syncthreads();
    }
    const float denom = s_red[0] + expf(se_i - m);
    __syncthreads();

    // ---- normalize into attention weights (adj value is 1.0 but keep it) ----
    for (int k = tid; k < cnt; k += 128)
        s_e[k] = expf(s_e[k] - m) * s_v[k] / denom;
    const float attn_self = expf(se_i - m) / denom;
    __syncthreads();

    // ---- aggregate: h = attn_self*Uq[i] + sum attn_k*Uk[j_k]; elu; store ----
    float h = attn_self * Uq[(size_t)i * 128 + tid];
    for (int k = 0; k < cnt; ++k)
        h = fmaf(s_e[k], Uk[(size_t)s_idx[k] * 128 + tid], h);
    out[(size_t)i * 128 + tid] = (h > 0.0f) ? h : (expf(h) - 1.0f);
}

// ---------------------------------------------------------------------------
extern "C" void kernel_launch(void* const* d_in, const int* in_sizes, int n_in,
                              void* d_out, int out_size, void* d_ws, size_t ws_size,
                              hipStream_t stream) {
    const int*   ver    = (const int*)d_in[0];
    const float* U9     = (const float*)d_in[1];   // 6000 x 128
    const float* U10    = (const float*)d_in[2];   // 9000 x 128
    const float* avec   = (const float*)d_in[3];   // 256 x 1
    const float* adj910 = (const float*)d_in[4];   // 6000 x 9000
    const float* adj109 = (const float*)d_in[5];   // 9000 x 6000
    float* out = (float*)d_out;

    float* ws    = (float*)d_ws;
    float* qa1   = ws;             // up to 9216 floats
    float* selfe = ws + 9216;
    float* ka2   = ws + 18432;

    const int N9 = 6000, N10 = 9000;
    const int tiles9  = (N9 + 15) / 16;    // 375
    const int tiles10 = (N10 + 15) / 16;   // 563
    const int blk9  = (tiles9 + 3) / 4;
    const int blk10 = (tiles10 + 3) / 4;

    // ---- version 0: query=U_9, key=U_10, adj=adj_9to10 ----
    gat_proj_wmma<<<blk9, 128, 0, stream>>>(ver, 0, U9, N9, avec, 0, qa1, selfe, ka2);
    gat_proj_wmma<<<blk10, 128, 0, stream>>>(ver, 0, U10, N10, avec, 1, qa1, selfe, ka2);
    gat_row<<<N9, 128, 0, stream>>>(ver, 0, U9, U10, adj910, qa1, selfe, ka2,
                                    N9, N10, out);

    // ---- version 1: query=U_10, key=U_9, adj=adj_10to9 ----
    gat_proj_wmma<<<blk10, 128, 0, stream>>>(ver, 1, U10, N10, avec, 0, qa1, selfe, ka2);
    gat_proj_wmma<<<blk9, 128, 0, stream>>>(ver, 1, U9, N9, avec, 1, qa1, selfe, ka2);
    gat_row<<<N10, 128, 0, stream>>>(ver, 1, U10, U9, adj109, qa1, selfe, ka2,
                                     N10, N9, out);
}